// Lowpass_56538949485209
// MI455X (gfx1250) — compile-verified
//
#include <hip/hip_runtime.h>
#include <cstdint>

#ifndef __has_builtin
#define __has_builtin(x) 0
#endif

#if __has_builtin(__builtin_amdgcn_global_load_async_to_lds_b128)
#define ASYNC_LDS 1
#endif

#define SRATE     44100.0f
#define BLK       128
#define SLEN      262144
#define NBATCH    32
#define NB        (SLEN / BLK)        /* 2048 blocks per batch   */
#define NBLOCKS   (NBATCH * NB)       /* 65536 filter blocks     */
#define WG        64                  /* 1 thread = 1 filter blk */
#define LSTRIDE   132                 /* 128+4 pad floats: 16B-aligned rows, low bank conflicts */
#define TILE_F    (WG * BLK)          /* 8192 floats per staged tile */
#define CHUNKS    (TILE_F / 4 / WG)   /* 32 float4 chunks per thread */

typedef float v4f __attribute__((ext_vector_type(4)));
typedef int   v4i __attribute__((ext_vector_type(4)));

#ifdef ASYNC_LDS
typedef __attribute__((address_space(1))) v4i GV4;   // global int4
typedef __attribute__((address_space(3))) v4i LV4;   // LDS int4
#endif

__device__ __forceinline__ void stage16(const float* g, float* l) {
#ifdef ASYNC_LDS
    // CDNA5 async copy: global -> LDS, tracked by ASYNCcnt, no VGPR data path.
    __builtin_amdgcn_global_load_async_to_lds_b128(
        (GV4*)(uintptr_t)g,
        (LV4*)(uint32_t)(uintptr_t)l,
        0, 0);
#else
    *(v4f*)l = *(const v4f*)g;
#endif
}

__device__ __forceinline__ void stage_wait() {
#ifdef ASYNC_LDS
#if __has_builtin(__builtin_amdgcn_s_wait_asynccnt)
    __builtin_amdgcn_s_wait_asynccnt(0);
#else
    asm volatile("s_wait_asynccnt 0" ::: "memory");
#endif
#endif
    __syncthreads();
}

// Copy TILE_F contiguous floats from gbase into the padded LDS tile,
// lane-consecutive float4 chunks -> fully coalesced global traffic.
__device__ __forceinline__ void stage_tile(const float* gbase, float* lds, int t) {
#pragma unroll
    for (int it = 0; it < CHUNKS; ++it) {
        int f = (t + it * WG) * 4;        // float index in tile (16B-aligned)
        int k = f >> 7;                   // which filter-block row
        stage16(gbase + f, lds + k * LSTRIDE + (f & (BLK - 1)));
    }
}

__global__ __launch_bounds__(WG) void lowpass_biquad_kernel(
    const float* __restrict__ x, const float* __restrict__ cp,
    float* __restrict__ y, float* __restrict__ fco, float* __restrict__ qo)
{
    __shared__ __align__(16) float tile[WG * LSTRIDE];   // 33 KB

    const int t  = threadIdx.x;
    const int g0 = blockIdx.x * WG;       // first filter block of this tile
    const int g  = g0 + t;                // this thread's filter block

    // x is (B,1,S): filter blocks are simply contiguous 128-float runs.
    const size_t xbase  = (size_t)g0 * BLK;
    // control is (B,2,S): channel c of the tile starts at batch*2*S + c*S + blk*128.
    const size_t cpbase = (size_t)(g0 >> 11) * 2u * SLEN
                        + (size_t)(g0 & (NB - 1)) * BLK;

    // Warm L2 for the x tile while we chew through control params.
    __builtin_prefetch(x + xbase + (size_t)t * BLK, 0, 1);

    float mean[2];
#pragma unroll
    for (int c = 0; c < 2; ++c) {
        stage_tile(cp + cpbase + (size_t)c * SLEN, tile, t);
        stage_wait();
        const v4f* row = (const v4f*)(tile + t * LSTRIDE);
        v4f acc = {0.f, 0.f, 0.f, 0.f};
#pragma unroll
        for (int i = 0; i < BLK / 4; ++i) acc += row[i];
        mean[c] = (acc.x + acc.y + acc.z + acc.w) * (1.0f / BLK);
        __syncthreads();                  // everyone done reading before re-stage
    }

    // Kick off the x-tile DMA, then overlap coefficient math with it.
    stage_tile(x + xbase, tile, t);

    const float fc = mean[0] * (20000.0f - 2000.0f) + 2000.0f;
    const float qv = mean[1] * (10.0f - 0.1f) + 0.1f;
    const float w0 = (2.0f * 3.14159265358979323846f / SRATE) * fc;
    float sw, cw;
    __sincosf(w0, &sw, &cw);
    const float alpha = sw / (2.0f * qv);
    const float a0inv = 1.0f / (1.0f + alpha);
    const float b1c = (1.0f - cw) * a0inv;
    const float b0c = 0.5f * b1c;
    const float b2c = b0c;
    const float a1c = (-2.0f * cw) * a0inv;
    const float a2c = (1.0f - alpha) * a0inv;

    stage_wait();

    // DF2T biquad over this thread's 128 samples, zero initial state.
    const float* xr = tile + t * LSTRIDE;
    float* yr = y + (size_t)g * BLK;
    float z1 = 0.f, z2 = 0.f;
#pragma unroll
    for (int i = 0; i < BLK; i += 4) {
        v4f o;
#pragma unroll
        for (int j = 0; j < 4; ++j) {
            float xn = xr[i + j];
            float yv = fmaf(b0c, xn, z1);
            z1 = fmaf(b1c, xn, fmaf(-a1c, yv, z2));
            z2 = fmaf(b2c, xn, -a2c * yv);
            o[j] = yv;
        }
        __builtin_nontemporal_store(o, (v4f*)(yr + i));   // streamed output: NT
    }
    __builtin_nontemporal_store(fc, fco + g);
    __builtin_nontemporal_store(qv, qo + g);
}

extern "C" void kernel_launch(void* const* d_in, const int* in_sizes, int n_in,
                              void* d_out, int out_size, void* d_ws, size_t ws_size,
                              hipStream_t stream) {
    (void)in_sizes; (void)n_in; (void)out_size; (void)d_ws; (void)ws_size;
    const float* x  = (const float*)d_in[0];
    const float* cp = (const float*)d_in[1];
    float* y  = (float*)d_out;                       // (32,1,262144)
    float* fc = y + (size_t)NBATCH * SLEN;           // (32,2048)
    float* q  = fc + NBLOCKS;                        // (32,2048)
    lowpass_biquad_kernel<<<NBLOCKS / WG, WG, 0, stream>>>(x, cp, y, fc, q);
}